// LinearModule_17188459118793
// MI455X (gfx1250) — compile-verified
//
#include <hip/hip_runtime.h>
#include <math.h>

// ---------------------------------------------------------------------------
// Problem constants (match reference): B=32768, L=50, D=50, R=3, C=10, OUT=1
// ---------------------------------------------------------------------------
#define L_DIM 50
#define D_DIM 50
#define R_DIM 3
#define C_DIM 10
#define FEAT  (R_DIM * D_DIM)   // 150
#define FPAD  152               // padded to multiple of 4 for K-chunking

#define WAVES_PER_BLOCK 4
#define PAIRS_PER_WAVE  16
#define PAIRS_PER_BLOCK (WAVES_PER_BLOCK * PAIRS_PER_WAVE)   // 64
#define ROWS_PER_WAVE   (PAIRS_PER_WAVE * R_DIM)             // 48

typedef __attribute__((ext_vector_type(2))) float v2f;
typedef __attribute__((ext_vector_type(8))) float v8f;

// D = A(16x4 f32) x B(4x16 f32) + C(16x16 f32), wave32.
// M = 16 (sample,combo) pairs per wave; K = feature chunks of 4; W in col N=0.
__global__ __launch_bounds__(WAVES_PER_BLOCK * 32)
void combo_dot_sigmoid_kernel(const int* __restrict__ x,          // [B*L]
                              const float* __restrict__ emb_mask, // [B*L]
                              const int* __restrict__ combo_idx,  // [B*C*R]
                              const float* __restrict__ emb_table,// [VOCAB*D]
                              const float* __restrict__ W,        // [FEAT]
                              const float* __restrict__ bias,     // [1]
                              float* __restrict__ out,            // [B]
                              int totalPairs)
{
    __shared__ float w_lds[FPAD];
    __shared__ float vec[WAVES_PER_BLOCK][PAIRS_PER_WAVE * FPAD]; // 16 pairs x 152 f32
    __shared__ int   tokS[WAVES_PER_BLOCK][ROWS_PER_WAVE];
    __shared__ float mskS[WAVES_PER_BLOCK][ROWS_PER_WAVE];

    const int tid      = threadIdx.x;
    const int wv       = tid >> 5;
    const int lane     = tid & 31;
    const int pairBase = blockIdx.x * PAIRS_PER_BLOCK + wv * PAIRS_PER_WAVE;

    // Weight vector -> LDS, zero-padded to FPAD.
    for (int i = tid; i < FPAD; i += blockDim.x)
        w_lds[i] = (i < FEAT) ? W[i] : 0.0f;

    // Resolve the indirection chain once per gathered row:
    // (pair, r) -> l = combo_idx -> token = x[sample][l], mask = emb_mask[sample][l]
    for (int row = lane; row < ROWS_PER_WAVE; row += 32) {
        int pr = row / R_DIM;
        int r  = row - pr * R_DIM;
        int pair = pairBase + pr;
        if (pair >= totalPairs) pair = totalPairs - 1;  // clamp; result discarded later
        int sample = pair / C_DIM;
        // pair == sample*C + c, so combo_idx[(sample*C+c)*R + r] == combo_idx[pair*R + r]
        int l = combo_idx[pair * R_DIM + r];
        tokS[wv][row] = x[sample * L_DIM + l];
        mskS[wv][row] = emb_mask[sample * L_DIM + l];
    }
    __syncthreads();

    // Stage masked embedding rows into LDS: 48 rows x 50 f32 (L2-resident gather).
    for (int idx = lane; idx < ROWS_PER_WAVE * D_DIM; idx += 32) {
        int row = idx / D_DIM;
        int d   = idx - row * D_DIM;
        int pr  = row / R_DIM;
        int r   = row - pr * R_DIM;
        float v = emb_table[(long long)tokS[wv][row] * D_DIM + d] * mskS[wv][row];
        vec[wv][pr * FPAD + r * D_DIM + d] = v;
    }
    // Zero the 2 pad features (150,151) of each of the 16 pairs: 32 slots, 1/lane.
    {
        int pr = lane >> 1;
        int f  = FEAT + (lane & 1);
        vec[wv][pr * FPAD + f] = 0.0f;
    }
    __syncthreads();

    // WMMA accumulation. ISA 16x4 f32 A layout: lane%16 = M; lanes 0-15 hold
    // K={0,1} in the two VGPRs, lanes 16-31 hold K={2,3}. B assumed mirrored:
    // lane%16 = N; lanes 0-15 hold K={0,1}, lanes 16-31 K={2,3}.
    const int m   = lane & 15;
    const int klo = (lane >> 4) << 1;                 // 0 or 2
    const float* vrow = &vec[wv][m * FPAD];
    v8f acc = {0.f, 0.f, 0.f, 0.f, 0.f, 0.f, 0.f, 0.f};
    for (int kc = 0; kc < FPAD / 4; ++kc) {           // 38 WMMAs
        int kb = kc * 4 + klo;
        v2f a;
        a.x = vrow[kb];
        a.y = vrow[kb + 1];
        float w0 = w_lds[kb], w1 = w_lds[kb + 1];
        v2f bm;
        bm.x = (m == 0) ? w0 : 0.0f;                  // W occupies column N=0 only
        bm.y = (m == 0) ? w1 : 0.0f;
        acc = __builtin_amdgcn_wmma_f32_16x16x4_f32(
            /*neg_a=*/false, a, /*neg_b=*/false, bm,
            /*c_mod=*/(short)0, acc, /*reuse_a=*/false, /*reuse_b=*/false);
    }

    // C/D layout: lane 0 holds (M=0..7, N=0) in acc[0..7]; lane 16 holds M=8..15.
    const float bval = bias[0];
    if ((lane & 15) == 0) {
        int mOff = (lane >> 4) * 8;
        #pragma unroll
        for (int v = 0; v < 8; ++v) {
            int pair = pairBase + mOff + v;
            if (pair < totalPairs) {
                float z = acc[v] + bval;
                float s = 1.0f / (1.0f + __expf(-z));
                atomicAdd(&out[pair / C_DIM], s);     // sum over the 10 combos
            }
        }
    }
}

// ---------------------------------------------------------------------------
// Inputs (setup_inputs order): 0:x(int) 1:emb_mask(f32) 2:combo_idx(int)
//                              3:step(int,unused) 4:emb_table(f32) 5:W(f32) 6:b(f32)
// Output: float [B] (OUT=1).
// ---------------------------------------------------------------------------
extern "C" void kernel_launch(void* const* d_in, const int* in_sizes, int n_in,
                              void* d_out, int out_size, void* d_ws, size_t ws_size,
                              hipStream_t stream) {
    const int*   x        = (const int*)d_in[0];
    const float* emb_mask = (const float*)d_in[1];
    const int*   combo    = (const int*)d_in[2];
    const float* table    = (const float*)d_in[4];
    const float* W        = (const float*)d_in[5];
    const float* b        = (const float*)d_in[6];
    float*       out      = (float*)d_out;

    const int Bsz        = in_sizes[0] / L_DIM;   // 32768
    const int totalPairs = Bsz * C_DIM;           // 327680 (multiple of 64)

    hipMemsetAsync(out, 0, (size_t)out_size * sizeof(float), stream);

    const int blocks = (totalPairs + PAIRS_PER_BLOCK - 1) / PAIRS_PER_BLOCK;
    combo_dot_sigmoid_kernel<<<blocks, WAVES_PER_BLOCK * 32, 0, stream>>>(
        x, emb_mask, combo, table, W, b, out, totalPairs);
}